// SparseMHA_78623671321107
// MI455X (gfx1250) — compile-verified
//
#include <hip/hip_runtime.h>
#include <hip/hip_bf16.h>

// ---------------------------------------------------------------------------
// SparseMHA (GAT-style) for gfx1250 / MI455X:
//   Kernel 1: fused Q/K/V projection GEMMs via V_WMMA_F32_16X16X4_F32,
//             async global->LDS staging of the A strip, dual-tile interleave.
//   Kernel 2: per-node edge scores + segment softmax + weighted aggregation.
// ---------------------------------------------------------------------------

#define NNODES 20000
#define DEG    16
#define HID    256
#define NH     8
#define HD     32
#define SCALEQ 0.17677669529663687f  // 1/sqrt(32)

typedef __attribute__((ext_vector_type(2))) float v2f;
typedef __attribute__((ext_vector_type(8))) float v8f;
typedef __attribute__((ext_vector_type(4))) int   v4i;

// Pointer types matching the async-LDS builtin prototype
// (param prints as "int __vector(4) __device__ *": v4i in AS1 / AS3).
typedef __attribute__((address_space(1))) v4i* gv4i_ptr;
typedef __attribute__((address_space(3))) v4i* lv4i_ptr;

// LDS row stride: multiple of 4 floats (16B alignment for b128 async stores),
// 260 % 64 == 4 -> each of the 16 rows starts on a different bank group.
#define LDA 260

#if defined(__gfx1250__) && __has_builtin(__builtin_amdgcn_global_load_async_to_lds_b128)
#define HAVE_ASYNC_LDS 1
#else
#define HAVE_ASYNC_LDS 0
#endif

#if defined(__gfx1250__) && __has_builtin(__builtin_amdgcn_s_wait_asynccnt)
#define WAIT_ASYNC() __builtin_amdgcn_s_wait_asynccnt(0)
#else
#define WAIT_ASYNC() asm volatile("s_wait_asynccnt 0" ::: "memory")
#endif

// ---------------------------------------------------------------------------
// Kernel 1: O[mat] = h @ W[mat] + b[mat]  (mat 0=q scaled, 1=k, 2=v)
// grid = N/16 blocks, 256 threads (8 waves).
// Wave w computes column tiles {w, w+8} for each of the 3 weight matrices,
// interleaving two independent WMMA accumulator chains.
// ---------------------------------------------------------------------------
__global__ __launch_bounds__(256)
void sparse_mha_qkv_wmma(const float* __restrict__ h,
                         const float* __restrict__ Wq, const float* __restrict__ bq,
                         const float* __restrict__ Wk, const float* __restrict__ bk,
                         const float* __restrict__ Wv, const float* __restrict__ bv,
                         float* __restrict__ qo, float* __restrict__ ko,
                         float* __restrict__ vo)
{
    __shared__ float As[16 * LDA];

    const int tid  = threadIdx.x;
    const int lane = tid & 31;
    const int wave = tid >> 5;          // 0..7
    const int row0 = blockIdx.x * 16;

    const int m  = lane & 15;           // A row / B,C,D column within tile
    const int kh = (lane >> 4) * 2;     // half-wave K selector (0 or 2)

    // Prefetch the first rows of each weight matrix for this wave's columns
    // while the A strip is being staged (emits global_prefetch_b8).
    {
        const int pc = wave * 16 + m;
        __builtin_prefetch(&Wq[pc], 0, 1);
        __builtin_prefetch(&Wk[pc], 0, 1);
        __builtin_prefetch(&Wv[pc], 0, 1);
    }

#if HAVE_ASYNC_LDS
    // Async stage: 16x256 floats = 1024 x b128 chunks, 4 per thread.
#pragma unroll
    for (int it = 0; it < 4; ++it) {
        const int j  = tid + it * 256;
        const int r  = j >> 6;            // 64 chunks per row
        const int c4 = (j & 63) << 2;
        const float* gsrc = &h[(size_t)(row0 + r) * HID + c4];
        float*       ldst = &As[r * LDA + c4];
        __builtin_amdgcn_global_load_async_to_lds_b128(
            (gv4i_ptr)gsrc, (lv4i_ptr)ldst, /*offset=*/0, /*cpol=*/0);
    }
    WAIT_ASYNC();
#else
    for (int i = tid; i < 16 * 256; i += 256) {
        const int r = i >> 8;
        const int c = i & 255;
        As[r * LDA + c] = h[(size_t)(row0 + r) * HID + c];
    }
#endif
    __syncthreads();

    const int col0 = wave * 16 + m;          // tiles 0..7
    const int col1 = (wave + 8) * 16 + m;    // tiles 8..15

#pragma unroll 1
    for (int p = 0; p < 3; ++p) {
        const float* __restrict__ W  = (p == 0) ? Wq : ((p == 1) ? Wk : Wv);
        const float* __restrict__ Bb = (p == 0) ? bq : ((p == 1) ? bk : bv);
        float* __restrict__       O  = (p == 0) ? qo : ((p == 1) ? ko : vo);

        // Fold bias into the accumulator (D = A*B + C with C = bias rows).
        const float bias0 = Bb[col0];
        const float bias1 = Bb[col1];
        v8f acc0, acc1;
#pragma unroll
        for (int r = 0; r < 8; ++r) { acc0[r] = bias0; acc1[r] = bias1; }

        for (int kk = 0; kk < HID; kk += 4) {
            const int ka = kk + kh;
            // Shared A fragment: A[m][ka], A[m][ka+1]  (16x4 f32 layout)
            const v2f afrag = *(const v2f*)&As[m * LDA + ka];
            // B fragments for two column tiles: B[ka][col], B[ka+1][col]
            const float* __restrict__ wrow = W + (size_t)ka * HID;
            v2f b0, b1;
            b0.x = wrow[col0];  b0.y = wrow[HID + col0];
            b1.x = wrow[col1];  b1.y = wrow[HID + col1];
            acc0 = __builtin_amdgcn_wmma_f32_16x16x4_f32(
                false, afrag, false, b0, (short)0, acc0, false, false);
            acc1 = __builtin_amdgcn_wmma_f32_16x16x4_f32(
                false, afrag, false, b1, (short)0, acc1, false, false);
        }

        const float scale = (p == 0) ? SCALEQ : 1.0f;
        const int   rbase = row0 + (lane >> 4) * 8;  // C/D: VGPR r -> M=r (+8 hi)
#pragma unroll
        for (int r = 0; r < 8; ++r) {
            O[(size_t)(rbase + r) * HID + col0] = acc0[r] * scale;
            O[(size_t)(rbase + r) * HID + col1] = acc1[r] * scale;
        }
    }
}

// ---------------------------------------------------------------------------
// Kernel 2: per-node scores, softmax over DEG=16 edges per head, aggregation.
// grid = N blocks, 128 threads each.
// ---------------------------------------------------------------------------
__global__ __launch_bounds__(128)
void sparse_mha_edge(const float* __restrict__ q, const float* __restrict__ k,
                     const float* __restrict__ v, const int* __restrict__ indices,
                     float* __restrict__ out)
{
    __shared__ float qs[HID];
    __shared__ float sc[DEG * NH];   // [edge][head]
    __shared__ int   nb[DEG];
    __shared__ float rdenom[NH];

    const int n = blockIdx.x;
    const int t = threadIdx.x;

    qs[t]       = q[(size_t)n * HID + t];
    qs[t + 128] = q[(size_t)n * HID + t + 128];
    if (t < DEG) nb[t] = indices[n * DEG + t];
    __syncthreads();

    // Scores: thread t -> (edge e, head h). Dot over 32 dims, stride NH.
    {
        const int e = t >> 3;
        const int h = t & 7;
        const float* __restrict__ kn = k + (size_t)nb[e] * HID;
        float s = 0.f;
#pragma unroll
        for (int d = 0; d < HD; ++d) {
            s += qs[d * NH + h] * kn[d * NH + h];   // q pre-scaled by 1/sqrt(HD)
        }
        sc[t] = s;
    }
    __syncthreads();

    // Softmax per head across the 16 edges (8 threads, one head each).
    if (t < NH) {
        float mx = -3.402823466e38f;
#pragma unroll
        for (int e = 0; e < DEG; ++e) mx = fmaxf(mx, sc[e * NH + t]);
        float dsum = 0.f;
#pragma unroll
        for (int e = 0; e < DEG; ++e) {
            const float ex = __expf(sc[e * NH + t] - mx);
            sc[e * NH + t] = ex;
            dsum += ex;
        }
        rdenom[t] = 1.0f / dsum;
    }
    __syncthreads();

    // Aggregation: thread t handles output columns t and t+128 (same head).
    const int h = t & 7;
    const float rd = rdenom[h];
    float acc0 = 0.f, acc1 = 0.f;
#pragma unroll 4
    for (int e = 0; e < DEG; ++e) {
        const float* __restrict__ vn = v + (size_t)nb[e] * HID;
        const float a = sc[e * NH + h] * rd;
        acc0 += a * vn[t];
        acc1 += a * vn[t + 128];
    }
    out[(size_t)n * HID + t]       = acc0;
    out[(size_t)n * HID + t + 128] = acc1;
}

// ---------------------------------------------------------------------------
extern "C" void kernel_launch(void* const* d_in, const int* in_sizes, int n_in,
                              void* d_out, int out_size, void* d_ws, size_t ws_size,
                              hipStream_t stream)
{
    (void)in_sizes; (void)n_in; (void)out_size; (void)ws_size;

    const float* h  = (const float*)d_in[0];
    const float* Wq = (const float*)d_in[1];
    const float* bq = (const float*)d_in[2];
    const float* Wk = (const float*)d_in[3];
    const float* bk = (const float*)d_in[4];
    const float* Wv = (const float*)d_in[5];
    const float* bv = (const float*)d_in[6];
    // d_in[7] = row (implicit for fixed degree), d_in[8] = indices
    const int* indices = (const int*)d_in[8];

    float* out = (float*)d_out;

    const size_t nhid = (size_t)NNODES * HID;
    float* qw = (float*)d_ws;
    float* kw = qw + nhid;
    float* vw = kw + nhid;

    sparse_mha_qkv_wmma<<<NNODES / 16, 256, 0, stream>>>(
        h, Wq, bq, Wk, bk, Wv, bv, qw, kw, vw);

    sparse_mha_edge<<<NNODES, 128, 0, stream>>>(qw, kw, vw, indices, out);
}